// PairEmbed_65712999629557
// MI455X (gfx1250) — compile-verified
//
#include <hip/hip_runtime.h>

typedef __attribute__((ext_vector_type(2))) float v2f;
typedef __attribute__((ext_vector_type(8))) float v8f;

#define EPSF     1e-8f
#define TWO_PI_F 6.28318530717958647692f

// One workgroup (256 threads = 8 wave32) per event (N = 4096).
// Phase 0: per-particle derived quantities + fragment-ready weights -> LDS
// Phase 1: conv1 as WMMA GEMM out1(8x256) = W1(8x64) x F(64x256),
//          F (pair features) precomputed per-lane in registers (8 pairs/lane/tile)
// Phase 2: conv2 (2x2 stride 2 pad 1) from zero-padded LDS -> global (N*8, 9, 9)
__global__ __launch_bounds__(256) void pair_embed_fused(
    const float* __restrict__ fm,   // (N, 4, 64)  E, px, py, pz
    const float* __restrict__ w1,   // (8, 4, 4, 4) = (8, 64)
    const float* __restrict__ b1,   // (8)
    const float* __restrict__ w2,   // (8, 8, 2, 2)
    const float* __restrict__ b2,   // (8)
    float*       __restrict__ out,  // (N*8, 9, 9)
    int N)
{
    __shared__ float sE[64], sPX[64], sPY[64], sPZ[64], sPT[64], sY[64], sPH[64];
    __shared__ float sW1A[1024];        // A-fragment layout: ((kq*2+half)*16+m)*2+e
    __shared__ float sW2[256];
    __shared__ float sB1[8], sB2[8];
    __shared__ float sO1p[8 * 324];     // conv1 output, padded 18x18 per channel

    const int tid = threadIdx.x;
    const int n   = blockIdx.x;

    // ---- Phase 0 ----
    // W1 in WMMA-A fragment order, zero-padded for rows m>=8:
    //   sW1A[kq*64 + half*32 + m*2 + e] = w1[m*64 + kq*4 + half*2 + e]
    #pragma unroll
    for (int s0 = 0; s0 < 1024; s0 += 256) {
        int s  = s0 + tid;
        int e  = s & 1;
        int mm = (s >> 1) & 15;
        int h  = (s >> 5) & 1;
        int kq = s >> 6;
        sW1A[s] = (mm < 8) ? w1[mm * 64 + kq * 4 + h * 2 + e] : 0.0f;
    }
    sW2[tid] = w2[tid];
    if (tid < 8) { sB1[tid] = b1[tid]; sB2[tid] = b2[tid]; }
    // zero padded conv1 output (border must be 0 for conv2)
    for (int s = tid; s < 8 * 324; s += 256) sO1p[s] = 0.0f;
    if (tid < 64) {
        const float* base = fm + (size_t)n * 256;
        float E  = base[tid];
        float px = base[64  + tid];
        float py = base[128 + tid];
        float pz = base[192 + tid];
        float pt = sqrtf(px * px + py * py + EPSF);
        float y  = 0.5f * logf((E + pz + EPSF) / (E - pz + EPSF));
        float ph = atan2f(py, px + EPSF);
        ph = ph - floorf(ph * (1.0f / TWO_PI_F)) * TWO_PI_F;   // jnp.mod(., 2pi)
        sE[tid] = E; sPX[tid] = px; sPY[tid] = py; sPZ[tid] = pz;
        sPT[tid] = pt; sY[tid] = y; sPH[tid] = ph;
    }
    __syncthreads();

    // ---- Phase 1: conv1 via V_WMMA_F32_16X16X4_F32 ----
    const int lane = tid & 31;
    const int wave = tid >> 5;
    const int m    = lane & 15;             // A row / col-in-tile
    const int half = lane >> 4;             // K half: lanes 0-15 -> K0,K1; 16-31 -> K2,K3
    const int koff = half * 2;

    // Preload all 16 A fragments (shared by both tiles of this wave).
    v2f aF[16];
    #pragma unroll
    for (int kq = 0; kq < 16; ++kq) {
        int idx = kq * 64 + half * 32 + m * 2;
        aF[kq] = *(const v2f*)&sW1A[idx];   // ds_load_b64, conflict-free banks
    }

    // j-side particle quantities: j = 4*m + koff + e  (tile-independent, 2 per lane)
    const int jbase = 4 * m + koff;
    float Ej[2], PXj[2], PYj[2], PZj[2], PTj[2], Yj[2], PHj[2];
    #pragma unroll
    for (int e = 0; e < 2; ++e) {
        int j = jbase + e;
        Ej[e] = sE[j];  PXj[e] = sPX[j]; PYj[e] = sPY[j]; PZj[e] = sPZ[j];
        PTj[e] = sPT[j]; Yj[e] = sY[j];  PHj[e] = sPH[j];
    }

    #pragma unroll
    for (int tt = 0; tt < 2; ++tt) {
        const int tile = wave * 2 + tt;     // 16 column tiles over 8 waves

        // Precompute all 4 channels for the 8 (i,j) pairs this lane feeds.
        float pf[4][4][2];                  // [channel][u][e], registers
        #pragma unroll
        for (int u = 0; u < 4; ++u) {
            const int i = tile * 4 + u;     // wave-uniform -> LDS broadcast reads
            const float Ei  = sE[i],  pxi = sPX[i], pyi = sPY[i], pzi = sPZ[i];
            const float pti = sPT[i], yi  = sY[i],  phi = sPH[i];
            #pragma unroll
            for (int e = 0; e < 2; ++e) {
                const bool off = (i != (jbase + e));
                float dy    = yi  - Yj[e];
                float dph   = phi - PHj[e];
                float delta = sqrtf(fabsf(dy * dy + dph * dph));
                float ptmin = fminf(pti, PTj[e]);
                float sx = pxi + PXj[e];
                float sy = pyi + PYj[e];
                float sz = pzi + PZj[e];
                float m2 = Ei * Ei + Ej[e] * Ej[e]
                         - (sx * sx + sy * sy + sz * sz);
                pf[0][u][e] = off ? delta : 0.0f;
                pf[1][u][e] = off ? (ptmin * delta) : 0.0f;
                pf[2][u][e] = off ? (ptmin / (pti + PTj[e] + EPSF)) : 0.0f;
                pf[3][u][e] = off ? m2 : 0.0f;
            }
        }

        // 16 chained WMMAs: B fragments are pure register references.
        v8f acc = {0.f, 0.f, 0.f, 0.f, 0.f, 0.f, 0.f, 0.f};
        #pragma unroll
        for (int kq = 0; kq < 16; ++kq) {
            const int c = kq >> 2;          // compile-time channel
            const int u = kq & 3;           // compile-time kernel row
            v2f b;
            b.x = pf[c][u][0];
            b.y = pf[c][u][1];
            acc = __builtin_amdgcn_wmma_f32_16x16x4_f32(
                false, aF[kq], false, b, (short)0, acc, false, false);
        }

        // D: lanes 0-15 hold rows M=0..7 in VGPRs 0..7. Store into padded 18x18.
        if (lane < 16) {
            const int base = (tile + 1) * 18 + (m + 1);   // (p+1, q+1)
            #pragma unroll
            for (int r = 0; r < 8; ++r)
                sO1p[r * 324 + base] = acc[r] + sB1[r];
        }
    }
    __syncthreads();

    // ---- Phase 2: conv2 (8,8,2,2), stride 2, pad 1 -> (8,9,9), branchless ----
    float* outn = out + (size_t)n * 648;
    for (int idx = tid; idx < 648; idx += 256) {
        int o  = idx / 81;
        int rs = idx - o * 81;
        int r  = rs / 9;
        int s  = rs - r * 9;
        float acc2 = sB2[o];
        #pragma unroll
        for (int c = 0; c < 8; ++c) {
            #pragma unroll
            for (int u = 0; u < 2; ++u) {
                #pragma unroll
                for (int v = 0; v < 2; ++v) {
                    acc2 += sW2[o * 32 + c * 4 + u * 2 + v]
                          * sO1p[c * 324 + (2 * r + u) * 18 + (2 * s + v)];
                }
            }
        }
        outn[idx] = acc2;
    }
}

extern "C" void kernel_launch(void* const* d_in, const int* in_sizes, int n_in,
                              void* d_out, int out_size, void* d_ws, size_t ws_size,
                              hipStream_t stream) {
    (void)n_in; (void)out_size; (void)d_ws; (void)ws_size;
    const float* fm = (const float*)d_in[0];   // (N,4,64)
    const float* w1 = (const float*)d_in[1];   // (8,4,4,4)
    const float* b1 = (const float*)d_in[2];   // (8)
    const float* w2 = (const float*)d_in[3];   // (8,8,2,2)
    const float* b2 = (const float*)d_in[4];   // (8)
    float* out = (float*)d_out;                // (N*8,9,9)
    int N = in_sizes[0] / 256;                 // 4096
    pair_embed_fused<<<N, 256, 0, stream>>>(fm, w1, b1, w2, b2, out, N);
}